// GAT_22170621181955
// MI455X (gfx1250) — compile-verified
//
#include <hip/hip_runtime.h>
#include <math.h>
#include <stdint.h>

typedef float v2f __attribute__((ext_vector_type(2)));
typedef float v8f __attribute__((ext_vector_type(8)));

#define N_NODES 10000
#define F_IN    256
#define D_OUT   128

// ---------------------------------------------------------------------------
// Kernel 0: h_prime = X @ W   ([10000,256] x [256,128], f32, WMMA 16x16x4)
// One wave per 16-row tile, covering all 128 output columns (8 C tiles).
// ---------------------------------------------------------------------------
__global__ __launch_bounds__(256) void gat_hprime_wmma(
    const float* __restrict__ X, const float* __restrict__ W,
    float* __restrict__ HP)
{
    const int lane   = threadIdx.x & 31;
    const int wave   = threadIdx.x >> 5;
    const int tile   = blockIdx.x * 8 + wave;
    if (tile * 16 >= N_NODES) return;          // wave-uniform guard

    const int  ln     = lane & 15;
    const int  hiHalf = lane >> 4;             // 0: lanes 0-15, 1: lanes 16-31
    const int  koff   = hiHalf * 2;            // K sub-offset per ISA A/B layout
    const long rowA   = (long)tile * 16 + ln;

    v8f c[8];
    #pragma unroll
    for (int dt = 0; dt < 8; ++dt)
        c[dt] = (v8f){0.f,0.f,0.f,0.f,0.f,0.f,0.f,0.f};

    for (int kk = 0; kk < F_IN; kk += 4) {
        v2f av;
        av.x = X[rowA * F_IN + kk + koff];
        av.y = X[rowA * F_IN + kk + koff + 1];
        #pragma unroll
        for (int dt = 0; dt < 8; ++dt) {
            const int col = dt * 16 + ln;
            v2f bv;
            bv.x = W[(long)(kk + koff)     * D_OUT + col];
            bv.y = W[(long)(kk + koff + 1) * D_OUT + col];
            c[dt] = __builtin_amdgcn_wmma_f32_16x16x4_f32(
                        false, av, false, bv, (short)0, c[dt], false, false);
        }
    }

    const int rbase = tile * 16 + hiHalf * 8;
    #pragma unroll
    for (int r = 0; r < 8; ++r) {
        const long row = rbase + r;
        #pragma unroll
        for (int dt = 0; dt < 8; ++dt)
            HP[row * D_OUT + dt * 16 + ln] = c[dt][r];
    }
}

// ---------------------------------------------------------------------------
// Kernel 0b: permute h_prime into the exact WMMA B-fragment register layout:
//   HPerm[((k4*32 + lane)*8 + dt)*2 + e] = HP[(4*k4 + koff(lane) + e)*128 + dt*16 + (lane&15)]
// so kernel 3 fetches all B-values a lane needs for one k-step as
// contiguous 16B-aligned chunks (global_load_b128).
// ---------------------------------------------------------------------------
__global__ __launch_bounds__(256) void gat_permute_b(
    const float* __restrict__ HP, float* __restrict__ HPerm)
{
    const int id = blockIdx.x * 256 + threadIdx.x;      // float2 index
    if (id >= (N_NODES / 4) * 32 * 8) return;
    const int dt   = id & 7;
    const int lane = (id >> 3) & 31;
    const int k4   = id >> 8;
    const int koff = (lane >> 4) * 2;
    const int col  = dt * 16 + (lane & 15);
    const int kb   = k4 * 4 + koff;
    float2 v;
    v.x = HP[(size_t)kb       * D_OUT + col];
    v.y = HP[(size_t)(kb + 1) * D_OUT + col];
    ((float2*)HPerm)[id] = v;
}

// ---------------------------------------------------------------------------
// Kernel 1: s_dst[j] = h_prime[j,:] . (W_a @ a[2:])  (s_src cancels in the
// row-wise softmax, so it is never needed)
// ---------------------------------------------------------------------------
__global__ __launch_bounds__(128) void gat_sdst(
    const float* __restrict__ HP, const float* __restrict__ Wa,
    const float* __restrict__ av, float* __restrict__ Sdst)
{
    __shared__ float red[128];
    const int row = blockIdx.x;
    const int t   = threadIdx.x;
    const float vd = Wa[t * 2 + 0] * av[2] + Wa[t * 2 + 1] * av[3];
    red[t] = HP[(long)row * D_OUT + t] * vd;
    __syncthreads();
    #pragma unroll
    for (int s = 64; s > 0; s >>= 1) {
        if (t < s) red[t] += red[t + s];
        __syncthreads();
    }
    if (t == 0) Sdst[row] = red[0];
}

// ---------------------------------------------------------------------------
// Kernel 2: per-row masked softmax stats: m_i, 1/Z_i (single coalesced
// Ahat pass, online max/sum).
// ---------------------------------------------------------------------------
__global__ __launch_bounds__(256) void gat_softmax_stats(
    const int* __restrict__ A, const float* __restrict__ Sdst,
    float* __restrict__ Mrow, float* __restrict__ InvZ)
{
    __shared__ float lm[256];
    __shared__ float ls[256];
    const int  row  = blockIdx.x;
    const int  t    = threadIdx.x;
    const long base = (long)row * N_NODES;

    float m = -3.0e38f, ssum = 0.0f;
    for (int j = t; j < N_NODES; j += 256) {
        if (A[base + j] > 0) {
            const float x = Sdst[j];
            if (x > m) { ssum = ssum * __expf(m - x) + 1.0f; m = x; }
            else       { ssum += __expf(x - m); }
        }
    }
    lm[t] = m; ls[t] = ssum;
    __syncthreads();
    #pragma unroll
    for (int s = 128; s > 0; s >>= 1) {
        if (t < s) {
            const float m1 = lm[t],     m2 = lm[t + s];
            const float s1 = ls[t],     s2 = ls[t + s];
            const float M  = fmaxf(m1, m2);
            lm[t] = M;
            ls[t] = s1 * __expf(m1 - M) + s2 * __expf(m2 - M);
        }
        __syncthreads();
    }
    if (t == 0) {
        Mrow[row] = lm[0];
        InvZ[row] = 1.0f / ls[0];
    }
}

// ---------------------------------------------------------------------------
// Kernel 3: out = relu(attn @ h_prime) — the 25.6 GFLOP GEMM.
// Block = 4 waves = 64 rows x 128 cols. Wave w owns column tiles {2w, 2w+1}
// for all 4 row-tiles (B read once per block). The 64x16 Ahat panel is
// DOUBLE-BUFFERED in LDS via GLOBAL_LOAD_ASYNC_TO_LDS_B128 (ASYNCcnt):
// panel n+1 streams from HBM while panel n feeds the matrix pipe.
// Attention A-fragments are built in-register:
//   A[i,j] = Ahat[i,j]>0 ? exp(s_dst[j]-m_0)*fac[rt] : 0,
// with fac[rt] = exp(m_0 - m_rt)/Z_rt folding per-row normalization so only
// 2 transcendentals are needed per lane per k-step.
// ---------------------------------------------------------------------------
#define LDS_ROW   20                 // 16 cols + 4 pad ints: 16B-aligned b128
                                     // writes, conflict-free ds_load_b64 reads
#define PANEL_INT (64 * LDS_ROW)     // ints per panel buffer

__global__ __launch_bounds__(128) void gat_attn_gemm_wmma(
    const int* __restrict__ A, const float* __restrict__ HPerm,
    const float* __restrict__ Sdst, const float* __restrict__ Mrow,
    const float* __restrict__ InvZ, float* __restrict__ Out)
{
    __shared__ __align__(16) int ap[2][PANEL_INT];

    const int lane    = threadIdx.x & 31;
    const int wave    = threadIdx.x >> 5;
    const int ln      = lane & 15;
    const int hiHalf  = lane >> 4;
    const int koff    = hiHalf * 2;
    const int dt0     = wave * 2;            // this wave's column-tile pair
    const int rowbase = blockIdx.x * 64;

    // Per-lane softmax stats for the 4 row-tiles (row index clamped for the
    // partial last block; those rows are computed but never stored).
    float mr0 = 0.0f;
    float fac[4];
    #pragma unroll
    for (int rt = 0; rt < 4; ++rt) {
        int gr = rowbase + rt * 16 + ln;
        if (gr > N_NODES - 1) gr = N_NODES - 1;
        const float m = Mrow[gr];
        const float z = InvZ[gr];
        if (rt == 0) { mr0 = m; fac[0] = z; }
        else         { fac[rt] = __expf(mr0 - m) * z; }
    }

    v8f c[4][2];
    #pragma unroll
    for (int rt = 0; rt < 4; ++rt)
        #pragma unroll
        for (int t2 = 0; t2 < 2; ++t2)
            c[rt][t2] = (v8f){0.f,0.f,0.f,0.f,0.f,0.f,0.f,0.f};

    const float4*  HP4     = (const float4*)HPerm;
    const unsigned ldsBase = (unsigned)(uintptr_t)&ap[0][0];

    // Async panel stage: 256 x 16B chunks, 2 per thread.
    const int cchunk0 = threadIdx.x;            // r = c>>2, jblk = c&3
    #define ISSUE_PANEL(KB, BUF)                                               \
        do {                                                                   \
            _Pragma("unroll")                                                  \
            for (int i = 0; i < 2; ++i) {                                      \
                const int cc   = cchunk0 + i * 128;                            \
                const int r    = cc >> 2;                                      \
                const int jblk = cc & 3;                                       \
                int grow = rowbase + r;                                        \
                if (grow > N_NODES - 1) grow = N_NODES - 1;                    \
                const int* gsrc = A + (size_t)grow * N_NODES + (KB) + jblk * 4;\
                const unsigned ldst = ldsBase +                                \
                    (unsigned)((BUF) * PANEL_INT + r * LDS_ROW + jblk * 4) * 4u;\
                asm volatile("global_load_async_to_lds_b128 %0, %1, off"       \
                             :: "v"(ldst), "v"(gsrc) : "memory");              \
            }                                                                  \
        } while (0)

    // Prologue: fetch panel 0, make it visible.
    ISSUE_PANEL(0, 0);
    asm volatile("s_wait_asynccnt 0" ::: "memory");
    __syncthreads();

    int cur = 0;
    for (int kb = 0; kb < N_NODES; kb += 16) {
        // Prefetch next panel into the other buffer while computing this one.
        if (kb + 16 < N_NODES) ISSUE_PANEL(kb + 16, cur ^ 1);

        const int* apc = &ap[cur][0];
        #pragma unroll
        for (int p4 = 0; p4 < 4; ++p4) {
            const int k4 = (kb >> 2) + p4;
            const int jl = p4 * 4 + koff;     // this lane's local K pair
            const float2 sd = *(const float2*)(Sdst + kb + jl);
            const float  e0 = __expf(sd.x - mr0);
            const float  e1 = __expf(sd.y - mr0);

            const float4 bq = HP4[(size_t)(k4 * 32 + lane) * 4 + (dt0 >> 1)];
            v2f b0; b0.x = bq.x; b0.y = bq.y;
            v2f b1; b1.x = bq.z; b1.y = bq.w;

            #pragma unroll
            for (int rt = 0; rt < 4; ++rt) {
                const int2 mk = *(const int2*)&apc[(rt * 16 + ln) * LDS_ROW + jl];
                v2f av;
                av.x = (mk.x > 0) ? e0 * fac[rt] : 0.0f;
                av.y = (mk.y > 0) ? e1 * fac[rt] : 0.0f;
                c[rt][0] = __builtin_amdgcn_wmma_f32_16x16x4_f32(
                               false, av, false, b0, (short)0, c[rt][0], false, false);
                c[rt][1] = __builtin_amdgcn_wmma_f32_16x16x4_f32(
                               false, av, false, b1, (short)0, c[rt][1], false, false);
            }
        }

        // Next panel complete (own loads) + all waves done with current one.
        asm volatile("s_wait_asynccnt 0" ::: "memory");
        __syncthreads();
        cur ^= 1;
    }
    #undef ISSUE_PANEL

    // Fused ReLU epilogue
    #pragma unroll
    for (int rt = 0; rt < 4; ++rt) {
        const int tile = blockIdx.x * 4 + rt;
        if (tile * 16 < N_NODES) {
            const int rb = tile * 16 + hiHalf * 8;
            #pragma unroll
            for (int t2 = 0; t2 < 2; ++t2) {
                const int col = (dt0 + t2) * 16 + ln;
                #pragma unroll
                for (int r = 0; r < 8; ++r)
                    Out[(size_t)(rb + r) * D_OUT + col] = fmaxf(c[rt][t2][r], 0.0f);
            }
        }
    }
}

// ---------------------------------------------------------------------------
extern "C" void kernel_launch(void* const* d_in, const int* in_sizes, int n_in,
                              void* d_out, int out_size, void* d_ws, size_t ws_size,
                              hipStream_t stream)
{
    const float* X  = (const float*)d_in[0];   // node_feats [10000,256]
    const int*   Ah = (const int*)  d_in[1];   // Ahat       [10000,10000]
    const float* W  = (const float*)d_in[2];   // w          [256,128]
    const float* Wa = (const float*)d_in[3];   // w_a        [128,2]
    const float* av = (const float*)d_in[4];   // a          [4,1]
    float* out = (float*)d_out;                // [10000,128]

    float* ws    = (float*)d_ws;
    float* HP    = ws;                 // 1,280,000 floats
    float* Sdst  = ws + 1280000;       // 10,000
    float* Mr    = ws + 1290000;       // 10,000
    float* Iz    = ws + 1300000;       // 10,000
    float* HPerm = ws + 1310000;       // 1,280,000  (total ~10.4 MB)

    gat_hprime_wmma   <<<79,      256, 0, stream>>>(X, W, HP);         // 625 tiles / 8 waves
    gat_permute_b     <<<2500,    256, 0, stream>>>(HP, HPerm);
    gat_sdst          <<<N_NODES, 128, 0, stream>>>(HP, Wa, av, Sdst);
    gat_softmax_stats <<<N_NODES, 256, 0, stream>>>(Ah, Sdst, Mr, Iz);
    gat_attn_gemm_wmma<<<157,     128, 0, stream>>>(Ah, HPerm, Sdst, Mr, Iz, out); // 625/4
}